// QuantumBlock_3092376453329
// MI455X (gfx1250) — compile-verified
//
#include <hip/hip_runtime.h>
#include <math.h>

typedef __attribute__((ext_vector_type(2))) float v2f;
typedef __attribute__((ext_vector_type(8))) float v8f;

#define LANE() ((int)(threadIdx.x & 31))

// branch-free tanh on v_exp_f32: saturates correctly for |x| large
__device__ __forceinline__ float fast_tanh(float x) {
  const float e = __expf(2.0f * x);
  return 1.0f - 2.0f / (e + 1.0f);
}

// ---------------------------------------------------------------------------
// WMMA f32 16x16x4 GEMM tile:  D(16 x 16) = A(16 x K) * B(K x 16) + bias
//   A operand layout (ISA 7.12.2, 32-bit A 16x4): lanes 0-15 -> K=k0,k0+1 ;
//   lanes 16-31 -> K=k0+2,k0+3 ; M = lane&15 in both halves.
//   B mirrors A with N across lanes.  C/D: VGPR r -> M = (lane>>4)*8 + r,
//   N = lane&15.
// ---------------------------------------------------------------------------
template<int K, int LDA, bool TANH>
__device__ __forceinline__ void gemm_tile_to_lds(
    const float* __restrict__ sA,        // 16 x K (LDS), row stride LDA
    const float* __restrict__ Wg,        // [N][K] row-major: B[k][n] = Wg[n*K+k]
    const float* __restrict__ bg,        // bias per output column
    float* __restrict__ sOut, int ldo, int n0)
{
  const int lane = LANE();
  const int half = lane >> 4;
  const int col  = lane & 15;
  const int n    = n0 + col;
  v8f acc = {};
#pragma unroll
  for (int k0 = 0; k0 < K; k0 += 4) {
    const int ka = k0 + half * 2;
    v2f a, b;
    a.x = sA[col * LDA + ka];
    a.y = sA[col * LDA + ka + 1];
    b.x = Wg[n * K + ka];
    b.y = Wg[n * K + ka + 1];
    acc = __builtin_amdgcn_wmma_f32_16x16x4_f32(false, a, false, b,
                                                (short)0, acc, false, false);
  }
  const float bias = bg[n];
#pragma unroll
  for (int r = 0; r < 8; ++r) {
    float v = acc[r] + bias;
    if (TANH) v = fast_tanh(v);
    sOut[(half * 8 + r) * ldo + n] = v;
  }
}

__device__ __forceinline__ void gemm_final(
    const float* __restrict__ sZ,        // 16 x 8 (LDS)
    const float* __restrict__ Wp,        // [128][8]
    const float* __restrict__ bp,
    float* __restrict__ out, int rowBase, int n0)
{
  const int lane = LANE();
  const int half = lane >> 4;
  const int col  = lane & 15;
  const int n    = n0 + col;
  v8f acc = {};
#pragma unroll
  for (int k0 = 0; k0 < 8; k0 += 4) {
    const int ka = k0 + half * 2;
    v2f a, b;
    a.x = sZ[col * 8 + ka];
    a.y = sZ[col * 8 + ka + 1];
    b.x = Wp[n * 8 + ka];
    b.y = Wp[n * 8 + ka + 1];
    acc = __builtin_amdgcn_wmma_f32_16x16x4_f32(false, a, false, b,
                                                (short)0, acc, false, false);
  }
  const float bias = bp[n];
#pragma unroll
  for (int r = 0; r < 8; ++r)
    out[(size_t)(rowBase + half * 8 + r) * 128 + n] = acc[r] + bias;
}

// ---------------------------------------------------------------------------
// 8-qubit state-vector simulator, one wave32 per sample.
// Flat amplitude index: bits [7:3] = lane id, bits [2:0] = local slot j.
// Wire w (PennyLane axis order, wire0 = MSB):  bit position = 7-w.
//   w <= 4  -> lane bit  LB = 4-w   (cross-lane, __shfl_xor)
//   w >= 5  -> local bit JB = 7-w   (in-register)
// ---------------------------------------------------------------------------
template<int JB>
__device__ __forceinline__ void gate_local(float ar[8], float ai[8],
    float u00r, float u00i, float u01r, float u01i,
    float u10r, float u10i, float u11r, float u11i)
{
#pragma unroll
  for (int j = 0; j < 8; ++j) {
    if (!((j >> JB) & 1)) {
      const int j1 = j | (1 << JB);
      const float a0r = ar[j],  a0i = ai[j];
      const float a1r = ar[j1], a1i = ai[j1];
      ar[j]  = u00r*a0r - u00i*a0i + u01r*a1r - u01i*a1i;
      ai[j]  = u00r*a0i + u00i*a0r + u01r*a1i + u01i*a1r;
      ar[j1] = u10r*a0r - u10i*a0i + u11r*a1r - u11i*a1i;
      ai[j1] = u10r*a0i + u10i*a0r + u11r*a1i + u11i*a1r;
    }
  }
}

template<int LB>
__device__ __forceinline__ void gate_lane(float ar[8], float ai[8],
    float u00r, float u00i, float u01r, float u01i,
    float u10r, float u10i, float u11r, float u11i)
{
  const int bit = (LANE() >> LB) & 1;
  const float csr = bit ? u11r : u00r;   // coefficient on self
  const float csi = bit ? u11i : u00i;
  const float cpr = bit ? u10r : u01r;   // coefficient on partner
  const float cpi = bit ? u10i : u01i;
#pragma unroll
  for (int j = 0; j < 8; ++j) {
    const float pr = __shfl_xor(ar[j], 1 << LB, 32);
    const float pi = __shfl_xor(ai[j], 1 << LB, 32);
    const float nr = csr*ar[j] - csi*ai[j] + cpr*pr - cpi*pi;
    const float ni = csr*ai[j] + csi*ar[j] + cpr*pi + cpi*pr;
    ar[j] = nr; ai[j] = ni;
  }
}

template<int W>
__device__ __forceinline__ void apply_gate(float ar[8], float ai[8],
    float u00r, float u00i, float u01r, float u01i,
    float u10r, float u10i, float u11r, float u11i)
{
  if constexpr (W <= 4)
    gate_lane<4 - W>(ar, ai, u00r, u00i, u01r, u01i, u10r, u10i, u11r, u11i);
  else
    gate_local<7 - W>(ar, ai, u00r, u00i, u01r, u01i, u10r, u10i, u11r, u11i);
}

// CNOT variants (pure permutations)
template<int CJB, int TJB>
__device__ __forceinline__ void cnot_ll(float ar[8], float ai[8]) {
#pragma unroll
  for (int j = 0; j < 8; ++j)
    if (((j >> CJB) & 1) && !((j >> TJB) & 1)) {
      const int j1 = j | (1 << TJB);
      float tr = ar[j], ti = ai[j];
      ar[j] = ar[j1]; ai[j] = ai[j1];
      ar[j1] = tr;    ai[j1] = ti;
    }
}
template<int CJB, int TLB>
__device__ __forceinline__ void cnot_local_lane(float ar[8], float ai[8]) {
#pragma unroll
  for (int j = 0; j < 8; ++j) {
    const float pr = __shfl_xor(ar[j], 1 << TLB, 32);
    const float pi = __shfl_xor(ai[j], 1 << TLB, 32);
    if ((j >> CJB) & 1) { ar[j] = pr; ai[j] = pi; }
  }
}
template<int CLB, int TJB>
__device__ __forceinline__ void cnot_lane_local(float ar[8], float ai[8]) {
  if ((LANE() >> CLB) & 1) {
#pragma unroll
    for (int j = 0; j < 8; ++j)
      if (!((j >> TJB) & 1)) {
        const int j1 = j | (1 << TJB);
        float tr = ar[j], ti = ai[j];
        ar[j] = ar[j1]; ai[j] = ai[j1];
        ar[j1] = tr;    ai[j1] = ti;
      }
  }
}
template<int CLB, int TLB>
__device__ __forceinline__ void cnot_lane_lane(float ar[8], float ai[8]) {
  const int cb = (LANE() >> CLB) & 1;
#pragma unroll
  for (int j = 0; j < 8; ++j) {
    const float pr = __shfl_xor(ar[j], 1 << TLB, 32);
    const float pi = __shfl_xor(ai[j], 1 << TLB, 32);
    if (cb) { ar[j] = pr; ai[j] = pi; }
  }
}

// Rot(phi,theta,omega) = RZ(omega) @ RY(theta) @ RZ(phi) -> u[8] =
// {u00r,u00i,u01r,u01i,u10r,u10i,u11r,u11i}.  Fast trig (v_sin/v_cos).
__device__ __forceinline__ void rot_u(float phi, float th, float om,
                                      float* __restrict__ u)
{
  const float pr = __cosf(0.5f * phi), pi = -__sinf(0.5f * phi); // e^{-i phi/2}
  const float qr = __cosf(0.5f * om),  qi = -__sinf(0.5f * om);  // e^{-i omega/2}
  const float cy = __cosf(0.5f * th),  sy =  __sinf(0.5f * th);
  const float qpr  = qr*pr - qi*pi, qpi  = qr*pi + qi*pr;        // q*p
  const float qcpr = qr*pr + qi*pi, qcpi = qi*pr - qr*pi;        // q*conj(p)
  u[0] =  cy*qpr;  u[1] =  cy*qpi;        // u00 = q  cy p
  u[2] = -sy*qcpr; u[3] = -sy*qcpi;       // u01 = -q sy p~
  u[4] =  sy*qcpr; u[5] = -sy*qcpi;       // u10 = q~ sy p  = sy*conj(q p~)
  u[6] =  cy*qpr;  u[7] = -cy*qpi;        // u11 = q~ cy p~ = cy*conj(q p)
}

#define INIT_WIRE(W) {                                                        \
    const float aY = ang[(W)];                                                \
    const float cc = __cosf(0.5f * aY), ss = __sinf(0.5f * aY);               \
    apply_gate<(W)>(ar, ai, cc,0.f, -ss,0.f, ss,0.f, cc,0.f);                 \
    const float aZ = ang[8 + (W)];                                            \
    const float zr = __cosf(0.5f * aZ), zi = -__sinf(0.5f * aZ);              \
    apply_gate<(W)>(ar, ai, zr,zi, 0.f,0.f, 0.f,0.f, zr,-zi);                 \
  }

#define ROT_WIRE(L, W) {                                                      \
    const float* u = &sU[((L)*8 + (W)) * 8];                                  \
    apply_gate<(W)>(ar, ai, u[0],u[1],u[2],u[3],u[4],u[5],u[6],u[7]);         \
  }

// ---------------------------------------------------------------------------
// Fused kernel: 512 threads = 16 wave32s, 16 batch rows per workgroup.
// ---------------------------------------------------------------------------
__global__ __launch_bounds__(512) void quantum_block_kernel(
    const float* __restrict__ x,
    const float* __restrict__ W1, const float* __restrict__ b1,
    const float* __restrict__ W2, const float* __restrict__ b2,
    const float* __restrict__ W3, const float* __restrict__ b3,
    const float* __restrict__ qw,
    const float* __restrict__ Wp, const float* __restrict__ bp,
    float* __restrict__ out)
{
  __shared__ float sX  [16 * 80];
  __shared__ float sH1 [16 * 64];
  __shared__ float sH2 [16 * 32];
  __shared__ float sAng[16 * 16];
  __shared__ float sZ  [16 * 8];
  __shared__ float sU  [24 * 8];   // 3 layers x 8 wires x complex 2x2

  const int tid     = threadIdx.x;
  const int wid     = tid >> 5;
  const int lane    = tid & 31;
  const int rowBase = blockIdx.x * 16;

  // Precompute the 24 batch-shared Rot matrices (threads 0..23), while all
  // threads also stage the x tile (16 x 80 contiguous floats).
  if (tid < 24) {
    const int L = tid >> 3, W = tid & 7;
    rot_u(qw[(L * 8 + W) * 3 + 0], qw[(L * 8 + W) * 3 + 1],
          qw[(L * 8 + W) * 3 + 2], &sU[tid * 8]);
  }
  for (int i = tid; i < 16 * 80; i += 512) sX[i] = x[rowBase * 80 + i];
  __syncthreads();

  // MLP: 80 -> 64 (tanh) -> 32 (tanh) -> 16 angles, all via f32 WMMA
  if (wid < 4) gemm_tile_to_lds<80, 80, true >(sX,  W1, b1, sH1, 64, wid * 16);
  __syncthreads();
  if (wid < 2) gemm_tile_to_lds<64, 64, true >(sH1, W2, b2, sH2, 32, wid * 16);
  __syncthreads();
  if (wid == 0) gemm_tile_to_lds<32, 32, false>(sH2, W3, b3, sAng, 16, 0);
  __syncthreads();

  // ---- quantum circuit: wave `wid` simulates sample rowBase + wid ----
  {
    float ar[8], ai[8];
#pragma unroll
    for (int j = 0; j < 8; ++j) { ar[j] = 0.f; ai[j] = 0.f; }
    if (lane == 0) ar[0] = 1.0f;           // |0...0>

    const float* ang = &sAng[wid * 16];
    INIT_WIRE(0) INIT_WIRE(1) INIT_WIRE(2) INIT_WIRE(3)
    INIT_WIRE(4) INIT_WIRE(5) INIT_WIRE(6) INIT_WIRE(7)

    for (int L = 0; L < 3; ++L) {
      ROT_WIRE(L, 0) ROT_WIRE(L, 1) ROT_WIRE(L, 2) ROT_WIRE(L, 3)
      ROT_WIRE(L, 4) ROT_WIRE(L, 5) ROT_WIRE(L, 6) ROT_WIRE(L, 7)
      cnot_lane_lane<4, 3>(ar, ai);   // CNOT(0,1)
      cnot_lane_lane<3, 2>(ar, ai);   // CNOT(1,2)
      cnot_lane_lane<2, 1>(ar, ai);   // CNOT(2,3)
      cnot_lane_lane<1, 0>(ar, ai);   // CNOT(3,4)
      cnot_lane_local<0, 2>(ar, ai);  // CNOT(7,2)
      cnot_ll<2, 1>(ar, ai);          // CNOT(5,6)
      cnot_ll<1, 0>(ar, ai);          // CNOT(6,7)
      cnot_local_lane<0, 4>(ar, ai);  // CNOT(7,0)
    }

    // <Z_w> = sum over amplitudes of (+/-)|a|^2
    float pb[8]; float tot = 0.f;
#pragma unroll
    for (int j = 0; j < 8; ++j) { pb[j] = ar[j]*ar[j] + ai[j]*ai[j]; tot += pb[j]; }
    float zp[8];
#pragma unroll
    for (int w = 0; w < 5; ++w)
      zp[w] = ((lane >> (4 - w)) & 1) ? -tot : tot;
    float s5 = 0.f, s6 = 0.f, s7 = 0.f;
#pragma unroll
    for (int j = 0; j < 8; ++j) {
      s5 += ((j >> 2) & 1) ? -pb[j] : pb[j];
      s6 += ((j >> 1) & 1) ? -pb[j] : pb[j];
      s7 += ( j       & 1) ? -pb[j] : pb[j];
    }
    zp[5] = s5; zp[6] = s6; zp[7] = s7;
#pragma unroll
    for (int w = 0; w < 8; ++w) {
      float v = zp[w];
#pragma unroll
      for (int off = 16; off > 0; off >>= 1) v += __shfl_xor(v, off, 32);
      if (lane == 0) sZ[wid * 8 + w] = v;
    }
  }
  __syncthreads();

  // readout projection: z(16x8) @ Wp^T(8x128) + bp, 8 waves x 16-col tiles
  if (wid < 8) gemm_final(sZ, Wp, bp, out, rowBase, wid * 16);
}

extern "C" void kernel_launch(void* const* d_in, const int* in_sizes, int n_in,
                              void* d_out, int out_size, void* d_ws, size_t ws_size,
                              hipStream_t stream) {
  const float* x  = (const float*)d_in[0];
  const float* W1 = (const float*)d_in[1];
  const float* b1 = (const float*)d_in[2];
  const float* W2 = (const float*)d_in[3];
  const float* b2 = (const float*)d_in[4];
  const float* W3 = (const float*)d_in[5];
  const float* b3 = (const float*)d_in[6];
  const float* qw = (const float*)d_in[7];
  const float* Wp = (const float*)d_in[8];
  const float* bp = (const float*)d_in[9];
  float* out = (float*)d_out;

  const int B = in_sizes[0] / 80;          // 16384
  const int grid = B / 16;                 // 16 samples per workgroup
  quantum_block_kernel<<<grid, 512, 0, stream>>>(x, W1, b1, W2, b2, W3, b3,
                                                 qw, Wp, bp, out);
}